// GCN_27625229648342
// MI455X (gfx1250) — compile-verified
//
#include <hip/hip_runtime.h>

#define NND   10000
#define NE    160000
#define HD    1280
#define INDIM 10
#define ODIM  7
#define NBLK  5
#define BNEPS 1e-5f

typedef float v2f __attribute__((ext_vector_type(2)));
typedef float v8f __attribute__((ext_vector_type(8)));

// gfx1250 async global->LDS path (ASYNCcnt-tracked), guarded so the file
// compiles on toolchains lacking the builtins. Builtin signature (from the
// compiler diagnostic): pointers to 16-byte int vectors, global src first.
#if __has_builtin(__builtin_amdgcn_global_load_async_to_lds_b128) && \
    __has_builtin(__builtin_amdgcn_s_wait_asynccnt)
#define ASYNC_LDS 1
typedef __attribute__((vector_size(16))) int vs4i;
#define ASYNC_CP16(gp, lp)                                          \
    __builtin_amdgcn_global_load_async_to_lds_b128(                 \
        (__attribute__((address_space(1))) vs4i*)(gp),              \
        (__attribute__((address_space(3))) vs4i*)(lp), 0, 0)
#define WAIT_ASYNC(n) __builtin_amdgcn_s_wait_asynccnt(n)
#else
#define ASYNC_LDS 0
#endif

// ---------------------------------------------------------------------------
// degree / normalization
// ---------------------------------------------------------------------------
__global__ void deg_init_kernel(float* deg, int n) {
    int i = blockIdx.x * blockDim.x + threadIdx.x;
    if (i < n) deg[i] = 1.0f;                       // self-loop contribution
}

__global__ void deg_accum_kernel(const int* __restrict__ ei, float* deg, int e) {
    int i = blockIdx.x * blockDim.x + threadIdx.x;
    if (i < e) atomicAdd(&deg[ei[NE + i]], 1.0f);   // dst side
}

__global__ void dinv_kernel(float* deg, int n) {
    int i = blockIdx.x * blockDim.x + threadIdx.x;
    if (i < n) deg[i] = rsqrtf(deg[i]);             // deg >= 1 always
}

// ---------------------------------------------------------------------------
// stem: h = relu(x @ W_stem + b_stem)   (K = 10, trivial)
// ---------------------------------------------------------------------------
__global__ __launch_bounds__(256) void stem_kernel(
    const float* __restrict__ x, const float* __restrict__ Ws,
    const float* __restrict__ bs, float* __restrict__ h) {
    int c = blockIdx.x * 256 + threadIdx.x;
    int node = blockIdx.y;
    float acc = bs[c];
    const float* xr = x + node * INDIM;
#pragma unroll
    for (int k = 0; k < INDIM; ++k)
        acc = fmaf(xr[k], Ws[k * HD + c], acc);
    h[(size_t)node * HD + c] = fmaxf(acc, 0.0f);
}

// ---------------------------------------------------------------------------
// WMMA f32 GEMM, 64(M) x 128(N) block tile, BK=32, double-buffered LDS with
// async global->LDS copies. 8 waves as 2(M) x 4(N); each wave owns a 32x32
// tile (4 v8f accumulators) -> 4 WMMAs per 4 LDS fragment reads per k-step.
// EPI=0: C = A@B                         (m = h @ Wg)
// EPI=1: Hres = BN(Hres + relu(A@B + b)) (fused GEMM + residual + BatchNorm)
// ---------------------------------------------------------------------------
#define LDA 36
#define LDB 132

template <int EPI>
__global__ __launch_bounds__(256) void gemm_wmma_kernel(
    const float* __restrict__ A,      // [nrows x K]
    const float* __restrict__ Bw,     // [K x HD]
    float* __restrict__ C,            // EPI=0 output
    const float* __restrict__ bias,
    const float* __restrict__ gma, const float* __restrict__ bta,
    const float* __restrict__ rmn, const float* __restrict__ rvr,
    float* __restrict__ Hres,         // EPI=1 residual in / out
    int nrows, int K) {

    __shared__ float lA[2][64 * LDA];
    __shared__ float lB[2][32 * LDB];

    const int tid    = threadIdx.x;
    const int lane   = tid & 31;
    const int wid    = tid >> 5;          // 0..7
    const int wN     = wid & 3;           // 4 waves along N (32 cols each)
    const int wM     = wid >> 2;          // 2 waves along M (32 rows each)
    const int laneLo = lane & 15;
    const int laneHi = lane >> 4;

    const int row0 = blockIdx.y * 64;     // M tile base
    const int col0 = blockIdx.x * 128;    // N tile base

    // A staging: 4 threads/row, 8 floats each (2 x b128)
    const int ar = tid >> 2;              // 0..63
    const int ak = (tid & 3) * 8;
    // B staging: 8 threads/k-row, 16 floats each (4 x b128)
    const int bk = tid >> 3;              // 0..31
    const int bn = (tid & 7) * 16;

    // Clamp OOB rows instead of zero-filling: garbage A rows only feed
    // C rows >= nrows, which the epilogue never stores.
    const int arow = min(row0 + ar, nrows - 1);
    const float* gA = A + (size_t)arow * K + ak;
    const float* gB = Bw + (size_t)bk * HD + col0 + bn;

    v8f acc[2][2] = {};

    const int nChunks = K / 32;

#if ASYNC_LDS
    auto issue = [&](int kc, int b) {
        const float* pa = gA + kc * 32;
        float* la = &lA[b][ar * LDA + ak];
        ASYNC_CP16(pa, la);
        ASYNC_CP16(pa + 4, la + 4);
        const float* pb = gB + (size_t)kc * 32 * HD;
        float* lb = &lB[b][bk * LDB + bn];
        ASYNC_CP16(pb, lb);
        ASYNC_CP16(pb + 4, lb + 4);
        ASYNC_CP16(pb + 8, lb + 8);
        ASYNC_CP16(pb + 12, lb + 12);
    };
    issue(0, 0);
#endif

    for (int kc = 0; kc < nChunks; ++kc) {
        const int b = kc & 1;

#if ASYNC_LDS
        // prefetch next chunk into the other buffer, then wait for current
        if (kc + 1 < nChunks) {
            issue(kc + 1, b ^ 1);
            WAIT_ASYNC(6);    // 6 newer ops outstanding; older 6 (chunk kc) done
        } else {
            WAIT_ASYNC(0);
        }
        __syncthreads();
#else
        // synchronous fallback: stage via VGPRs into buffer b
        {
            const float* pa = gA + kc * 32;
            const float4 a0 = *(const float4*)(pa);
            const float4 a1 = *(const float4*)(pa + 4);
            const float* pb = gB + (size_t)kc * 32 * HD;
            const float4 b0 = *(const float4*)(pb);
            const float4 b1 = *(const float4*)(pb + 4);
            const float4 b2 = *(const float4*)(pb + 8);
            const float4 b3 = *(const float4*)(pb + 12);
            __syncthreads();   // previous compute done before overwrite
            *(float4*)(&lA[b][ar * LDA + ak])     = a0;
            *(float4*)(&lA[b][ar * LDA + ak + 4]) = a1;
            float* lb = &lB[b][bk * LDB + bn];
            *(float4*)(lb)      = b0;
            *(float4*)(lb + 4)  = b1;
            *(float4*)(lb + 8)  = b2;
            *(float4*)(lb + 12) = b3;
        }
        __syncthreads();
#endif

        // ---- compute: 8 k-steps of 4; 2 M-subtiles x 2 N-subtiles ----
#pragma unroll
        for (int ks = 0; ks < 8; ++ks) {
            const int kb = ks * 4 + 2 * laneHi;   // per-lane K base in chunk
            v2f bf[2], af[2];
#pragma unroll
            for (int nt = 0; nt < 2; ++nt) {
                const int cl = wN * 32 + nt * 16 + laneLo;
                bf[nt].x = lB[b][kb * LDB + cl];
                bf[nt].y = lB[b][(kb + 1) * LDB + cl];
            }
#pragma unroll
            for (int mt = 0; mt < 2; ++mt) {
                const int rl = wM * 32 + mt * 16 + laneLo;
                af[mt].x = lA[b][rl * LDA + kb];
                af[mt].y = lA[b][rl * LDA + kb + 1];
            }
#pragma unroll
            for (int mt = 0; mt < 2; ++mt)
#pragma unroll
                for (int nt = 0; nt < 2; ++nt)
                    acc[mt][nt] = __builtin_amdgcn_wmma_f32_16x16x4_f32(
                        false, af[mt], false, bf[nt], (short)0, acc[mt][nt],
                        false, false);
        }
#if ASYNC_LDS
        __syncthreads();      // all waves done with buffer b before it refills
#endif
    }

    // ---- epilogue ----
#pragma unroll
    for (int nt = 0; nt < 2; ++nt) {
        const int col = col0 + wN * 32 + nt * 16 + laneLo;
        float bv = 0.f, sc = 0.f, mu = 0.f, be = 0.f;
        if (EPI == 1) {
            bv = bias[col];
            sc = gma[col] * rsqrtf(rvr[col] + BNEPS);
            mu = rmn[col];
            be = bta[col];
        }
#pragma unroll
        for (int mt = 0; mt < 2; ++mt) {
#pragma unroll
            for (int r = 0; r < 8; ++r) {
                const int row = row0 + wM * 32 + mt * 16 + r + 8 * laneHi;
                if (row < nrows) {
                    float v = acc[mt][nt][r];
                    if (EPI == 0) {
                        C[(size_t)row * HD + col] = v;
                    } else {
                        v = fmaxf(v + bv, 0.0f);
                        float hv = Hres[(size_t)row * HD + col] + v;
                        Hres[(size_t)row * HD + col] = (hv - mu) * sc + be;
                    }
                }
            }
        }
    }
}

// ---------------------------------------------------------------------------
// aggregation: self-loop init, edge scatter-add, bias+relu
// ---------------------------------------------------------------------------
__global__ __launch_bounds__(256) void selfloop_kernel(
    const float* __restrict__ m, float* __restrict__ agg,
    const float* __restrict__ dinv) {
    int c = blockIdx.x * 256 + threadIdx.x;
    int node = blockIdx.y;
    float w = dinv[node];
    agg[(size_t)node * HD + c] = m[(size_t)node * HD + c] * w * w;
}

__global__ __launch_bounds__(256) void scatter_kernel(
    const float* __restrict__ m, float* __restrict__ agg,
    const int* __restrict__ ei, const float* __restrict__ dinv) {
    const int e = blockIdx.x;
    const int s = ei[e];
    const int d = ei[NE + e];
    const float w = dinv[s] * dinv[d];
    const float* ms = m + (size_t)s * HD;
    float* ad = agg + (size_t)d * HD;
    for (int c = threadIdx.x; c < HD; c += 256)
        atomicAdd(&ad[c], ms[c] * w);
}

__global__ __launch_bounds__(256) void bias_relu_kernel(
    float* __restrict__ agg, const float* __restrict__ bg) {
    int c = blockIdx.x * 256 + threadIdx.x;
    int node = blockIdx.y;
    size_t idx = (size_t)node * HD + c;
    agg[idx] = fmaxf(agg[idx] + bg[c], 0.0f);
}

// ---------------------------------------------------------------------------
// output: out = h @ W_out + b_out   (one wave32 per node, 7 accumulators)
// ---------------------------------------------------------------------------
__global__ __launch_bounds__(256) void out_kernel(
    const float* __restrict__ h, const float* __restrict__ Wo,
    const float* __restrict__ bo, float* __restrict__ out, int n) {
    const int gw = (blockIdx.x * 256 + threadIdx.x) >> 5;
    const int lane = threadIdx.x & 31;
    if (gw >= n) return;
    float acc[ODIM] = {};
    const float* hr = h + (size_t)gw * HD;
    for (int c = lane; c < HD; c += 32) {
        const float hv = hr[c];
#pragma unroll
        for (int o = 0; o < ODIM; ++o)
            acc[o] = fmaf(hv, Wo[c * ODIM + o], acc[o]);
    }
#pragma unroll
    for (int off = 16; off > 0; off >>= 1)
#pragma unroll
        for (int o = 0; o < ODIM; ++o)
            acc[o] += __shfl_down(acc[o], off, 32);
    if (lane == 0)
#pragma unroll
        for (int o = 0; o < ODIM; ++o)
            out[(size_t)gw * ODIM + o] = acc[o] + bo[o];
}

// ---------------------------------------------------------------------------
// launcher
// ---------------------------------------------------------------------------
extern "C" void kernel_launch(void* const* d_in, const int* in_sizes, int n_in,
                              void* d_out, int out_size, void* d_ws, size_t ws_size,
                              hipStream_t stream) {
    const float* x      = (const float*)d_in[0];
    const int*   ei     = (const int*)d_in[1];    // [2, E]
    const float* W_stem = (const float*)d_in[2];
    const float* b_stem = (const float*)d_in[3];
    const float* Wg     = (const float*)d_in[4];  // [NB,H,H]
    const float* bg     = (const float*)d_in[5];
    const float* Wl     = (const float*)d_in[6];
    const float* bl     = (const float*)d_in[7];
    const float* gamma_ = (const float*)d_in[8];
    const float* beta_  = (const float*)d_in[9];
    const float* rmean_ = (const float*)d_in[10];
    const float* rvar_  = (const float*)d_in[11];
    const float* W_out  = (const float*)d_in[12];
    const float* b_out  = (const float*)d_in[13];
    float* out = (float*)d_out;

    const size_t NH = (size_t)NND * HD;
    float* dinv = (float*)d_ws;                   // N floats
    float* h    = dinv + ((NND + 255) & ~255);    // N*H
    float* m    = h + NH;                         // N*H
    float* agg  = m + NH;                         // N*H

    deg_init_kernel<<<(NND + 255) / 256, 256, 0, stream>>>(dinv, NND);
    deg_accum_kernel<<<(NE + 255) / 256, 256, 0, stream>>>(ei, dinv, NE);
    dinv_kernel<<<(NND + 255) / 256, 256, 0, stream>>>(dinv, NND);

    stem_kernel<<<dim3(HD / 256, NND), 256, 0, stream>>>(x, W_stem, b_stem, h);

    const dim3 gemm_grid(HD / 128, (NND + 63) / 64);
    const dim3 elem_grid(HD / 256, NND);

    for (int i = 0; i < NBLK; ++i) {
        gemm_wmma_kernel<0><<<gemm_grid, 256, 0, stream>>>(
            h, Wg + (size_t)i * HD * HD, m,
            nullptr, nullptr, nullptr, nullptr, nullptr, nullptr, NND, HD);
        selfloop_kernel<<<elem_grid, 256, 0, stream>>>(m, agg, dinv);
        scatter_kernel<<<NE, 256, 0, stream>>>(m, agg, ei, dinv);
        bias_relu_kernel<<<elem_grid, 256, 0, stream>>>(agg, bg + (size_t)i * HD);
        gemm_wmma_kernel<1><<<gemm_grid, 256, 0, stream>>>(
            agg, Wl + (size_t)i * HD * HD, nullptr,
            bl + (size_t)i * HD, gamma_ + (size_t)i * HD, beta_ + (size_t)i * HD,
            rmean_ + (size_t)i * HD, rvar_ + (size_t)i * HD, h, NND, HD);
    }

    out_kernel<<<(NND * 32 + 255) / 256, 256, 0, stream>>>(h, W_out, b_out, out, NND);
}